// DifferentiableTree_36051955483049
// MI455X (gfx1250) — compile-verified
//
#include <hip/hip_runtime.h>
#include <stdint.h>

#define LNK 24
#define TPB 256
#define BPT 2   // batch elements (independent chains) per thread -> VOPD dual-issue ILP

// Forward kinematics for a serial revolute chain (B x 24 links -> B x 24 x 4x4).
// Store-bandwidth bound: 201 MB of output vs ~0.35 GFLOP of math.
// - Link constants staged into LDS once per block via CDNA5 async global->LDS
//   loads (ASYNCcnt), normalized in place.
// - Each thread runs BPT independent chains in registers so the compiler can
//   dual-issue FMAs across chains (the single-chain 27-FMA matmul is a serial
//   dependency otherwise).
__global__ __launch_bounds__(TPB) void fk_chain_kernel(
    const float* __restrict__ q,     // [B, LNK]
    const float* __restrict__ axes,  // [LNK, 3] (unnormalized)
    const float* __restrict__ tfix,  // [LNK, 3]
    float* __restrict__ out,         // [B, LNK, 4, 4]
    int B)
{
    // s_lc[0 .. 71]   = axes (normalized in place below)
    // s_lc[72 .. 143] = t_fix
    __shared__ float s_lc[6 * LNK];

    const int t = threadIdx.x;

    // ---- Stage link constants into LDS with async global->LDS loads ----
    if (t < 6 * LNK) {
        const float* src = (t < 3 * LNK) ? (axes + t) : (tfix + (t - 3 * LNK));
        uint32_t lds_addr = (uint32_t)(size_t)(void*)s_lc + (uint32_t)(t * 4);
        uint64_t gaddr    = (uint64_t)(uintptr_t)src;
        asm volatile("global_load_async_to_lds_b32 %0, %1, off"
                     :
                     : "v"(lds_addr), "v"(gaddr)
                     : "memory");
    }
    __builtin_amdgcn_s_wait_asynccnt(0);
    __syncthreads();

    // ---- Normalize the axes once (24 lanes of the block) ----
    if (t < LNK) {
        float a0 = s_lc[3 * t + 0];
        float a1 = s_lc[3 * t + 1];
        float a2 = s_lc[3 * t + 2];
        float inv = 1.0f / sqrtf(a0 * a0 + a1 * a1 + a2 * a2);
        s_lc[3 * t + 0] = a0 * inv;
        s_lc[3 * t + 1] = a1 * inv;
        s_lc[3 * t + 2] = a2 * inv;
    }
    __syncthreads();

    const int base = blockIdx.x * (TPB * BPT) + t;

    int    bidx[BPT];
    bool   act[BPT];
    float  qq[BPT][LNK];
    float4* o[BPT];

#pragma unroll
    for (int c = 0; c < BPT; ++c) {
        bidx[c] = base + c * TPB;           // lanes stay consecutive -> coalesced
        act[c]  = bidx[c] < B;
        o[c]    = (float4*)out + (size_t)bidx[c] * (LNK * 4);
        if (act[c]) {
            const float4* qv = (const float4*)(q + (size_t)bidx[c] * LNK);
#pragma unroll
            for (int i = 0; i < LNK / 4; ++i) {
                float4 v = qv[i];
                qq[c][4 * i + 0] = v.x;
                qq[c][4 * i + 1] = v.y;
                qq[c][4 * i + 2] = v.z;
                qq[c][4 * i + 3] = v.w;
            }
        }
    }

    // Chain state per sub-batch: R = I, t = 0
    float R[BPT][9];
    float T[BPT][3];
#pragma unroll
    for (int c = 0; c < BPT; ++c) {
        R[c][0] = 1.f; R[c][1] = 0.f; R[c][2] = 0.f;
        R[c][3] = 0.f; R[c][4] = 1.f; R[c][5] = 0.f;
        R[c][6] = 0.f; R[c][7] = 0.f; R[c][8] = 1.f;
        T[c][0] = 0.f; T[c][1] = 0.f; T[c][2] = 0.f;
    }

    for (int l = 0; l < LNK; ++l) {
        // Wave-uniform link constants (one broadcast read feeds both chains)
        const float a0 = s_lc[3 * l + 0];
        const float a1 = s_lc[3 * l + 1];
        const float a2 = s_lc[3 * l + 2];
        const float f0 = s_lc[3 * LNK + 3 * l + 0];
        const float f1 = s_lc[3 * LNK + 3 * l + 1];
        const float f2 = s_lc[3 * LNK + 3 * l + 2];

#pragma unroll
        for (int c = 0; c < BPT; ++c) {
            float sn, cs;
            __sincosf(qq[c][l], &sn, &cs);
            const float oc = 1.0f - cs;

            // Rodrigues: J = c*I + (1-c)*a*a^T + s*skew(a)
            const float xs0 = sn * a0, xs1 = sn * a1, xs2 = sn * a2;
            const float p01 = oc * a0 * a1;
            const float p02 = oc * a0 * a2;
            const float p12 = oc * a1 * a2;
            const float J00 = oc * a0 * a0 + cs;
            const float J11 = oc * a1 * a1 + cs;
            const float J22 = oc * a2 * a2 + cs;
            const float J01 = p01 - xs2, J02 = p02 + xs1;
            const float J10 = p01 + xs2, J12 = p12 - xs0;
            const float J20 = p02 - xs1, J21 = p12 + xs0;

            // t_child = t_parent + R_parent * t_fix   (uses OLD R)
            T[c][0] += R[c][0] * f0 + R[c][1] * f1 + R[c][2] * f2;
            T[c][1] += R[c][3] * f0 + R[c][4] * f1 + R[c][5] * f2;
            T[c][2] += R[c][6] * f0 + R[c][7] * f1 + R[c][8] * f2;

            // R_child = R_parent * J
            const float N0 = R[c][0] * J00 + R[c][1] * J10 + R[c][2] * J20;
            const float N1 = R[c][0] * J01 + R[c][1] * J11 + R[c][2] * J21;
            const float N2 = R[c][0] * J02 + R[c][1] * J12 + R[c][2] * J22;
            const float N3 = R[c][3] * J00 + R[c][4] * J10 + R[c][5] * J20;
            const float N4 = R[c][3] * J01 + R[c][4] * J11 + R[c][5] * J21;
            const float N5 = R[c][3] * J02 + R[c][4] * J12 + R[c][5] * J22;
            const float N6 = R[c][6] * J00 + R[c][7] * J10 + R[c][8] * J20;
            const float N7 = R[c][6] * J01 + R[c][7] * J11 + R[c][8] * J21;
            const float N8 = R[c][6] * J02 + R[c][7] * J12 + R[c][8] * J22;
            R[c][0] = N0; R[c][1] = N1; R[c][2] = N2;
            R[c][3] = N3; R[c][4] = N4; R[c][5] = N5;
            R[c][6] = N6; R[c][7] = N7; R[c][8] = N8;

            // Emit 4x4 pose: [R | t ; 0 0 0 1]  (64B fully-dirty aligned block)
            if (act[c]) {
                o[c][0] = make_float4(R[c][0], R[c][1], R[c][2], T[c][0]);
                o[c][1] = make_float4(R[c][3], R[c][4], R[c][5], T[c][1]);
                o[c][2] = make_float4(R[c][6], R[c][7], R[c][8], T[c][2]);
                o[c][3] = make_float4(0.f, 0.f, 0.f, 1.f);
                o[c] += 4;
            }
        }
    }
}

extern "C" void kernel_launch(void* const* d_in, const int* in_sizes, int n_in,
                              void* d_out, int out_size, void* d_ws, size_t ws_size,
                              hipStream_t stream) {
    const float* q    = (const float*)d_in[0];
    // d_in[1] = qd : unused by the reference output
    const float* axes = (const float*)d_in[2];
    const float* tfix = (const float*)d_in[3];
    float* out = (float*)d_out;

    const int B = in_sizes[0] / LNK;
    const int grid = (B + TPB * BPT - 1) / (TPB * BPT);
    fk_chain_kernel<<<grid, TPB, 0, stream>>>(q, axes, tfix, out, B);
}